// CF2DConv_15796889715292
// MI455X (gfx1250) — compile-verified
//
#include <hip/hip_runtime.h>

#define NX 512
#define NY 512
#define DV 32
#define NB 4
#define KX 64
#define KY 64

#define PI_D 3.14159265358979323846

typedef float v2f __attribute__((ext_vector_type(2)));
typedef float v8f __attribute__((ext_vector_type(8)));

// ---------------------------------------------------------------------------
// Basis generation: Adct (64x512), Fc/Fs (64x512), Gc/Gs (512x64).
//  Adct[b][n] = s_b cos(pi*b*(2n+1)/(2*NY)),   s_0=sqrt(1/NY), s_b=sqrt(2/NY)
//  Fc[a][m]   =  cos(2*pi*a*m/NX)/sqrt(NX)     (rfft ortho, real part)
//  Fs[a][m]   = -sin(2*pi*a*m/NX)/sqrt(NX)     (rfft ortho, imag part)
//  Gc[m][a]   =  w_a cos(2*pi*a*m/NX)/sqrt(NX) (irfft ortho, w_0=1, w_a=2)
//  Gs[m][a]   = -w_a sin(2*pi*a*m/NX)/sqrt(NX) (Gs[m][0]=0 -> Im(Y0) ignored)
// Phases reduced mod period exactly in integers, then double-precision trig.
// ---------------------------------------------------------------------------
__global__ void init_basis(float* __restrict__ Adct, float* __restrict__ Fc,
                           float* __restrict__ Fs, float* __restrict__ Gc,
                           float* __restrict__ Gs) {
  int idx = blockIdx.x * blockDim.x + threadIdx.x;
  if (idx >= KY * NY) return;

  const double invSqrtN = 0.044194173824159220;  // 1/sqrt(512)

  // DCT-II (ortho) truncated basis, 64x512
  {
    int b = idx >> 9, n = idx & 511;
    double s = (b == 0) ? sqrt(1.0 / (double)NY) : sqrt(2.0 / (double)NY);
    int t = (int)(((long)b * (2 * n + 1)) % (4L * NY));  // cos period: 4*NY
    Adct[idx] = (float)(s * cos(PI_D * (double)t / (2.0 * (double)NY)));
  }
  // rFFT (ortho) truncated basis, 64x512
  {
    int a = idx >> 9, m = idx & 511;
    int t = (int)(((long)a * m) & (NX - 1));
    double ang = (2.0 * PI_D / (double)NX) * (double)t;
    Fc[idx] = (float)(cos(ang) * invSqrtN);
    Fs[idx] = (float)(-sin(ang) * invSqrtN);
  }
  // irFFT (ortho, zero-padded beyond a=63) basis, 512x64
  {
    int m = idx >> 6, a = idx & 63;
    int t = (int)(((long)a * m) & (NX - 1));
    double ang = (2.0 * PI_D / (double)NX) * (double)t;
    double w = (a == 0) ? 1.0 : 2.0;
    Gc[idx] = (float)(w * cos(ang) * invSqrtN);
    Gs[idx] = (float)(-w * sin(ang) * invSqrtN);
  }
}

// ---------------------------------------------------------------------------
// Batched f32 WMMA GEMM, all shapes/strides compile-time:
//   C[M,N] (+)= A[M,K] * B[K,N]
//   A indexed A[r*RSA + k*CSA] (RSA=1,CSA=ld gives transposed basis access)
//   blockIdx.y = batch with element strides SA/SB/SC.
//   Each wave32 computes a 16 x (16*NBLK) strip: one A fragment feeds NBLK
//   independent V_WMMA_F32_16X16X4_F32 accumulator chains (ILP + A reuse).
//   Wave-uniform tile guard keeps EXEC all-ones inside WMMA (ISA req).
// ---------------------------------------------------------------------------
template <int M, int N, int K, int RSA, int CSA, int LDB, int LDC,
          long SA, long SB, long SC, int NBLK, bool ACC>
__global__ void gemm_wmma_f32(const float* __restrict__ A,
                              const float* __restrict__ B,
                              float* __restrict__ C) {
  constexpr int TN = (N / 16) / NBLK;     // col-tile groups per row of tiles
  constexpr int TILES = (M / 16) * TN;
  const int wave = (int)((blockIdx.x * blockDim.x + threadIdx.x) >> 5);
  if (wave >= TILES) return;  // uniform per-wave
  const int lane = (int)(threadIdx.x & 31);
  const long batch = blockIdx.y;
  const int tm = (wave / TN) * 16;
  const int tn = (wave % TN) * (16 * NBLK);
  const int half = lane >> 4;  // 0: lanes 0-15, 1: lanes 16-31
  const int l16 = lane & 15;

  const float* __restrict__ Ab = A + batch * SA + (long)(tm + l16) * RSA;
  const float* __restrict__ Bb = B + batch * SB + tn + l16;
  float* __restrict__ Cb = C + batch * SC + tn + l16;

  v8f acc[NBLK];
#pragma unroll
  for (int t = 0; t < NBLK; ++t) {
    if constexpr (ACC) {
#pragma unroll
      for (int v = 0; v < 8; ++v)
        acc[t][v] = Cb[(long)(tm + v + 8 * half) * LDC + 16 * t];
    } else {
      acc[t] = (v8f){0.f, 0.f, 0.f, 0.f, 0.f, 0.f, 0.f, 0.f};
    }
  }

#pragma unroll 4
  for (int k0 = 0; k0 < K; k0 += 4) {
    const int ka = k0 + 2 * half;  // V0 holds K = ka, V1 holds K = ka+1
    v2f af;
    if constexpr (CSA == 1) {
      af = *(const v2f*)(Ab + ka);  // 8B-aligned: rows even-strided, ka even
    } else {
      af.x = Ab[(long)ka * CSA];
      af.y = Ab[(long)(ka + 1) * CSA];
    }
    v2f bf[NBLK];
#pragma unroll
    for (int t = 0; t < NBLK; ++t) {
      bf[t].x = Bb[(long)ka * LDB + 16 * t];
      bf[t].y = Bb[(long)(ka + 1) * LDB + 16 * t];
    }
#pragma unroll
    for (int t = 0; t < NBLK; ++t)
      acc[t] = __builtin_amdgcn_wmma_f32_16x16x4_f32(
          false, af, false, bf[t], (short)0, acc[t], false, false);
  }

#pragma unroll
  for (int t = 0; t < NBLK; ++t)
#pragma unroll
    for (int v = 0; v < 8; ++v)
      Cb[(long)(tm + v + 8 * half) * LDC + 16 * t] = acc[t][v];
}

// ---------------------------------------------------------------------------
// Per-frequency complex channel mixing:  y[e,i] = sum_j R[i,j]*x[e,j]
// One block per (a,b) of the 64x64 retained modes; 128 threads = (e,i).
// R tile (32x32 complex) staged through LDS; X/Y layout [e][a][b*32 + ch].
// ---------------------------------------------------------------------------
__global__ void mode_mix(const float* __restrict__ Rre, const float* __restrict__ Rim,
                         const float* __restrict__ Xre, const float* __restrict__ Xim,
                         float* __restrict__ Yre, float* __restrict__ Yim) {
  const int bb = (int)blockIdx.x;  // b (fast dim -> adjacent blocks share R lines)
  const int aa = (int)blockIdx.y;  // a
  __shared__ float sRre[DV * DV];
  __shared__ float sRim[DV * DV];
  __shared__ float sXre[NB * DV];
  __shared__ float sXim[NB * DV];

  const int tid = (int)threadIdx.x;  // 128
  for (int idx = tid; idx < DV * DV; idx += 128) {
    const long off = ((long)idx * (KX * KY)) + (long)aa * KY + bb;  // R[i,j,a,b]
    sRre[idx] = Rre[off];
    sRim[idx] = Rim[off];
  }
  {
    const int e = tid >> 5, j = tid & 31;
    const long off = (long)e * (KX * KY * DV) + (long)aa * (KY * DV) + bb * DV + j;
    sXre[tid] = Xre[off];
    sXim[tid] = Xim[off];
  }
  __syncthreads();

  const int e = tid >> 5, i = tid & 31;
  float yre = 0.f, yim = 0.f;
#pragma unroll 8
  for (int j = 0; j < DV; ++j) {
    const float rr = sRre[i * DV + j], ri = sRim[i * DV + j];
    const float xr = sXre[e * DV + j], xi = sXim[e * DV + j];
    yre = fmaf(rr, xr, fmaf(-ri, xi, yre));
    yim = fmaf(rr, xi, fmaf(ri, xr, yim));
  }
  const long off = (long)e * (KX * KY * DV) + (long)aa * (KY * DV) + bb * DV + i;
  Yre[off] = yre;
  Yim[off] = yim;
}

// ---------------------------------------------------------------------------
// Pipeline (all f32, memory-bound: ~480 MB traffic, ~11 GFLOP):
//   A: T1[e,m][b*32+j]     = Adct(64x512) @ x[e,m](512x32)      [2048 slices]
//   B: Xre/Xim[e](64x2048) = Fc/Fs(64x512) @ T1[e](512x2048)    [4 batches]
//   C: per-mode complex mix -> Yre/Yim                          [4096 blocks]
//   D: yr[e](512x2048)     = Gc@Yre[e] + Gs@Yim[e]              [4 batches]
//   E: out[e,m](512x32)    = Adct^T(512x64) @ yr[e,m](64x32)    [2048 slices]
// ---------------------------------------------------------------------------
extern "C" void kernel_launch(void* const* d_in, const int* in_sizes, int n_in,
                              void* d_out, int out_size, void* d_ws, size_t ws_size,
                              hipStream_t stream) {
  const float* x = (const float*)d_in[0];
  const float* Rre = (const float*)d_in[1];
  const float* Rim = (const float*)d_in[2];
  float* out = (float*)d_out;
  float* ws = (float*)d_ws;

  float* Adct = ws;              // 64*512   = 32768
  float* Fc = ws + 32768;        // 64*512
  float* Fs = ws + 65536;        // 64*512
  float* Gc = ws + 98304;        // 512*64
  float* Gs = ws + 131072;       // 512*64
  float* T1 = ws + 163840;       // 4*512*2048 = 4194304  (reused as yr)
  float* Xre = T1 + 4194304;     // 4*64*2048 = 524288
  float* Xim = Xre + 524288;
  float* Yre = Xim + 524288;
  float* Yim = Yre + 524288;     // total ~25.8 MB of workspace

  init_basis<<<dim3(128), dim3(256), 0, stream>>>(Adct, Fc, Fs, Gc, Gs);

  // Stage A: truncated DCT along NY. M=64,N=32,K=512, NBLK=2 -> 4 waves/slice.
  gemm_wmma_f32<64, 32, 512, 512, 1, 32, 32, 0L, 16384L, 2048L, 2, false>
      <<<dim3(1, 2048), dim3(128), 0, stream>>>(Adct, x, T1);

  // Stage B: truncated rFFT along NX (real & imag). M=64,N=2048,K=512, NBLK=4.
  // tiles = 4*32 = 128 -> grid.x = 16 at 8 waves/block.
  gemm_wmma_f32<64, 2048, 512, 512, 1, 2048, 2048, 0L, 1048576L, 131072L, 4, false>
      <<<dim3(16, 4), dim3(256), 0, stream>>>(Fc, T1, Xre);
  gemm_wmma_f32<64, 2048, 512, 512, 1, 2048, 2048, 0L, 1048576L, 131072L, 4, false>
      <<<dim3(16, 4), dim3(256), 0, stream>>>(Fs, T1, Xim);

  // Stage C: per-mode complex 32x32 channel mixing.
  mode_mix<<<dim3(KY, KX), dim3(128), 0, stream>>>(Rre, Rim, Xre, Xim, Yre, Yim);

  // Stage D: zero-padded inverse rFFT along NX. yr = Gc@Yre + Gs@Yim.
  // M=512,N=2048,K=64, NBLK=4: tiles = 32*32 = 1024 -> grid.x = 128.
  gemm_wmma_f32<512, 2048, 64, 64, 1, 2048, 2048, 0L, 131072L, 1048576L, 4, false>
      <<<dim3(128, 4), dim3(256), 0, stream>>>(Gc, Yre, T1);
  gemm_wmma_f32<512, 2048, 64, 64, 1, 2048, 2048, 0L, 131072L, 1048576L, 4, true>
      <<<dim3(128, 4), dim3(256), 0, stream>>>(Gs, Yim, T1);

  // Stage E: inverse DCT along NY (Adct^T via RSA=1,CSA=512).
  // M=512,N=32,K=64, NBLK=2: tiles = 32 -> grid.x = 4 at 8 waves/block.
  gemm_wmma_f32<512, 32, 64, 1, 512, 32, 32, 0L, 2048L, 16384L, 2, false>
      <<<dim3(4, 2048), dim3(256), 0, stream>>>(Adct, T1, out);
}